// TorchModalitySampler_62431644614852
// MI455X (gfx1250) — compile-verified
//
#include <hip/hip_runtime.h>
#include <cstdint>

// Greedy peak-picking sampler for MI455X (gfx1250, wave32).
// One 1024-thread workgroup (32 wave32 waves) per batch image (the 6 greedy
// steps are serially dependent, so the batch is the natural parallel unit).
//
// CDNA5 paths:
//  - global_load_async_to_lds_b128 / global_store_async_from_lds_b128
//    (+ s_wait_asynccnt) for the input -> mutable-workspace DMA
//  - V_WMMA_F32_16X16X4_F32: horizontal 17-tap box sums as a banded-ones
//    matmul, 16 output rows per barrier pair (8x less LDS traffic, 16x fewer
//    barriers than the scalar 17-reads-per-output formulation)
//  - global_prefetch_b8 on the streamed rows, wave32 shuffle reductions.

namespace {
constexpr int W    = 1024;
constexpr int H    = 1024;
constexpr int NB   = 32;
constexpr int REC  = 17;            // 2*RADIUS+1
constexpr int AGGW = W - REC + 1;   // 1008 = 63*16
constexpr int AGGH = H - REC + 1;   // 1008 = 63*16
constexpr int NT   = 6;             // N_TARGETS
constexpr int BS   = 1024;          // threads per block (32 waves)
constexpr int GR   = 16;            // rows per group (WMMA M)
constexpr int NGRP = AGGH / GR;     // 63
constexpr int NTIL = AGGW / 16;     // 63 column tiles
constexpr int VP   = 1025;          // padded LDS row stride (bank de-alias)
constexpr float KINV = 1.0f / 289.0f;  // box-average scale (monotone)
}

typedef float v2f __attribute__((ext_vector_type(2)));
typedef float v8f __attribute__((ext_vector_type(8)));

#if defined(__AMDGCN__) && __has_builtin(__builtin_amdgcn_wmma_f32_16x16x4_f32)
#define HAVE_WMMA_F32X4 1
#else
#define HAVE_WMMA_F32X4 0
#endif

// Branchless (value, index) argmax update with first-flat-index tie-break.
// Non-short-circuit booleans -> v_cmp + v_cndmask, no EXEC manipulation.
__device__ __forceinline__ void upd_best(float a, int idx, float& best, int& bestIdx) {
  const bool better = (a > best) | ((a == best) & (idx < bestIdx));
  best    = better ? a   : best;
  bestIdx = better ? idx : bestIdx;
}

__global__ __launch_bounds__(BS)
void greedy_sampler_kernel(const float* __restrict__ src,
                           float* __restrict__ ws,
                           float* __restrict__ out) {
  const int b    = blockIdx.x;
  const int t    = threadIdx.x;
  const int lane = t & 31;
  const int wid  = t >> 5;

  const float* __restrict__ gsrc = src + (size_t)b * W * H;
  float* __restrict__ hm = ws + (size_t)b * W * H;

  __shared__ float s_vs[GR][VP];      // 16 vertical-sum rows (also DMA stage)
  __shared__ float s_rv[32];
  __shared__ float s_rv2[32];
  __shared__ int   s_ri[32];
  __shared__ int   s_rc[2];           // (r, c) of selected window
  __shared__ float s_stats[2];        // (window max, window sum)

  // ---------- Phase A: copy heatmap -> workspace via CDNA5 async LDS DMA ----
  {
    float* stage = &s_vs[0][0];                               // reuse as stage
    unsigned ldsoff = (unsigned)(uintptr_t)(&stage[t * 4]);   // LDS byte offset
    for (int it = 0; it < (W * H) / (BS * 4); ++it) {
      unsigned long long ga =
          (unsigned long long)(uintptr_t)(gsrc + it * BS * 4 + t * 4);
      asm volatile("global_load_async_to_lds_b128 %0, %1, off"
                   :: "v"(ldsoff), "v"(ga) : "memory");
      asm volatile("s_wait_asynccnt 0" ::: "memory");
      unsigned long long gd =
          (unsigned long long)(uintptr_t)(hm + it * BS * 4 + t * 4);
      asm volatile("global_store_async_from_lds_b128 %0, %1, off"
                   :: "v"(gd), "v"(ldsoff) : "memory");
      asm volatile("s_wait_asynccnt 0" ::: "memory");
    }
  }
  __syncthreads();

#if HAVE_WMMA_F32X4
  // Banded-ones B tiles for the horizontal 17-tap sum: B[k][n] = (0 <= d <= 16)
  // with d = 4*kt + k - n.  Pattern depends only on kt -> 8 constant fragments.
  // B(4x16) lane layout: N = lane&15, K = vgpr + 2*(lane>=16).
  v2f btile[8];
  {
    const int n  = lane & 15;
    const int kh = (lane >> 4) << 1;          // 0 or 2
    #pragma unroll
    for (int kt = 0; kt < 8; ++kt) {
      #pragma unroll
      for (int v = 0; v < 2; ++v) {
        const int d = 4 * kt + (kh + v) - n;
        btile[kt][v] = (d >= 0 && d <= REC - 1) ? 1.0f : 0.0f;
      }
    }
  }
#endif

  for (int step = 0; step < NT; ++step) {
    // ---------- Phase B: argmax over 17x17 box-average ----------
    float best    = -__builtin_inff();
    int   bestIdx = 0x7fffffff;

    for (int g = 0; g < NGRP; ++g) {
      const int r0 = g * GR;
      // Vertical 17-row sums for 16 consecutive output rows, column = t.
      // Fresh 17-term sum each group start bounds FP drift; then slide.
      float vs = 0.f;
      #pragma unroll
      for (int dr = 0; dr < REC; ++dr) vs += hm[(r0 + dr) * W + t];
      s_vs[0][t] = vs;
      #pragma unroll
      for (int i = 1; i < GR; ++i) {
        vs += hm[(r0 + i + REC - 1) * W + t] - hm[(r0 + i - 1) * W + t];
        s_vs[i][t] = vs;
      }
      if (r0 + 2 * GR < H)
        __builtin_prefetch(&hm[(r0 + 2 * GR) * W + t], 0, 1);
      __syncthreads();

#if HAVE_WMMA_F32X4
      // AGG tile (16x16) = sum_{kt<8} A(16x4) x B(4x16), A from s_vs band.
      // A lane layout: M = lane&15, K = vgpr + 2*(lane>=16).
      const int m  = lane & 15;
      const int kh = (lane >> 4) << 1;
      const int mbase = (lane >> 4) << 3;     // D: M = vgpr + 8*(lane>=16)
      for (int tile = wid; tile < NTIL; tile += 32) {   // wave-uniform
        const int c0 = tile * 16;
        v8f acc = {};
        #pragma unroll
        for (int kt = 0; kt < 8; ++kt) {
          v2f a;
          a[0] = s_vs[m][c0 + 4 * kt + kh + 0];
          a[1] = s_vs[m][c0 + 4 * kt + kh + 1];
          acc = __builtin_amdgcn_wmma_f32_16x16x4_f32(
              false, a, false, btile[kt], (short)0, acc, false, false);
        }
        #pragma unroll
        for (int v = 0; v < 8; ++v) {
          const float a = acc[v] * KINV;      // monotone box-average scale
          const int idx = (r0 + mbase + v) * AGGW + (c0 + m);
          upd_best(a, idx, best, bestIdx);
        }
      }
#else
      // Scalar fallback: 17-tap LDS sums, one column per thread.
      if (t < AGGW) {
        #pragma unroll
        for (int i = 0; i < GR; ++i) {
          float s = 0.f;
          #pragma unroll
          for (int dc = 0; dc < REC; ++dc) s += s_vs[i][t + dc];
          upd_best(s * KINV, (r0 + i) * AGGW + t, best, bestIdx);
        }
      }
#endif
      __syncthreads();
    }

    // Block-wide argmax, smaller-flat-index tie-break (first-occurrence).
    #pragma unroll
    for (int off = 16; off; off >>= 1) {
      float ov = __shfl_down(best, off);
      int   oi = __shfl_down(bestIdx, off);
      upd_best(ov, oi, best, bestIdx);
    }
    if (lane == 0) { s_rv[wid] = best; s_ri[wid] = bestIdx; }
    __syncthreads();
    if (wid == 0) {
      best = s_rv[lane]; bestIdx = s_ri[lane];
      #pragma unroll
      for (int off = 16; off; off >>= 1) {
        float ov = __shfl_down(best, off);
        int   oi = __shfl_down(bestIdx, off);
        upd_best(ov, oi, best, bestIdx);
      }
      if (lane == 0) { s_rc[0] = bestIdx / AGGW; s_rc[1] = bestIdx % AGGW; }
    }
    __syncthreads();
    const int rr = s_rc[0], cc = s_rc[1];

    // ---------- Phase C1: window max + sum (confidence) ----------
    const int dr = t / REC, dc = t % REC;
    float wmax = -__builtin_inff(), wsum = 0.f;
    if (t < REC * REC) {
      float v = hm[(rr + dr) * W + (cc + dc)];
      wmax = v; wsum = v;
    }
    #pragma unroll
    for (int off = 16; off; off >>= 1) {
      wmax  = fmaxf(wmax, __shfl_down(wmax, off));
      wsum += __shfl_down(wsum, off);
    }
    if (lane == 0) { s_rv[wid] = wmax; s_rv2[wid] = wsum; }
    __syncthreads();
    if (wid == 0) {
      wmax = s_rv[lane]; wsum = s_rv2[lane];
      #pragma unroll
      for (int off = 16; off; off >>= 1) {
        wmax  = fmaxf(wmax, __shfl_down(wmax, off));
        wsum += __shfl_down(wsum, off);
      }
      if (lane == 0) { s_stats[0] = wmax; s_stats[1] = wsum; }
    }
    __syncthreads();
    const float mv = s_stats[0];

    // ---------- Phase C2: first flat index in WHOLE hm equal to max_val ----
    // (replicates jnp.argmax(hm == max_val) first-occurrence semantics)
    int firstIdx = 0x7fffffff;
    for (int i = t; i < W * H; i += BS) {
      if (hm[i] == mv) { firstIdx = i; break; }  // ascending: first hit is min
    }
    #pragma unroll
    for (int off = 16; off; off >>= 1) {
      int o = __shfl_down(firstIdx, off);
      firstIdx = (o < firstIdx) ? o : firstIdx;
    }
    if (lane == 0) s_ri[wid] = firstIdx;
    __syncthreads();
    if (wid == 0) {
      firstIdx = s_ri[lane];
      #pragma unroll
      for (int off = 16; off; off >>= 1) {
        int o = __shfl_down(firstIdx, off);
        firstIdx = (o < firstIdx) ? o : firstIdx;
      }
      if (lane == 0) {
        out[(b * NT + step) * 2 + 0] = (float)(firstIdx / W); // row (UPSCALE=1)
        out[(b * NT + step) * 2 + 1] = (float)(firstIdx % W); // col
        out[NB * NT * 2 + b * NT + step] = s_stats[1];        // confidence
      }
    }
    __syncthreads();   // all scans complete before suppression

    // ---------- Phase D: zero the 17x17 window ----------
    if (t < REC * REC) hm[(rr + dr) * W + (cc + dc)] = 0.f;
    __syncthreads();   // visible to next step's Phase B (block-scope fence)
  }
}

extern "C" void kernel_launch(void* const* d_in, const int* in_sizes, int n_in,
                              void* d_out, int out_size, void* d_ws, size_t ws_size,
                              hipStream_t stream) {
  (void)in_sizes; (void)n_in; (void)out_size; (void)ws_size;
  const float* heatmap = (const float*)d_in[0];  // [32,1,1024,1024] f32
  float* out = (float*)d_out;                    // 384 pts + 192 confs
  float* ws  = (float*)d_ws;                     // needs 128 MB
  hipLaunchKernelGGL(greedy_sampler_kernel, dim3(NB), dim3(BS), 0, stream,
                     heatmap, ws, out);
}